// GeneralTV1DLayer_23536420782357
// MI455X (gfx1250) — compile-verified
//
#include <hip/hip_runtime.h>
#include <cstdint>
#include <cstddef>

// ---------------------------------------------------------------------------
// TV-1D prox (FISTA dual, 30 iters) for x:(8,32,128,256) f32, lambda per channel.
// One wave32 per 256-length row; 8 elements per lane held in VGPRs; neighbor
// exchange via wave32 shuffles (ds_bpermute); row staged global->LDS with the
// CDNA5 async-to-LDS engine (ASYNCcnt / s_wait_asynccnt).
// Shapes are fixed by the reference: (8, 32, 128, 256).
// ---------------------------------------------------------------------------

#define TV_N               256
#define TV_R_SHIFT         7     // rows per (batch,channel) slab = 128 = 1<<7
#define TV_C_MASK          31    // 32 channels
#define TV_ITERS           30
#define TV_WAVES_PER_BLOCK 8
#define TV_BLOCK           (TV_WAVES_PER_BLOCK * 32)

__global__ __launch_bounds__(TV_BLOCK) void tv1d_fista_kernel(
    const float* __restrict__ x,
    const float* __restrict__ lmbd_param,
    float* __restrict__ out,
    int nrows)
{
    __shared__ __align__(16) float smem[TV_WAVES_PER_BLOCK * TV_N]; // 8 KB

    const int lane = threadIdx.x & 31;
    const int wv   = threadIdx.x >> 5;
    const int row  = blockIdx.x * TV_WAVES_PER_BLOCK + wv;
    if (row >= nrows) return;            // whole wave exits together

    // per-channel lambda = softplus(raw)  (numerically stable form);
    // channel index = (row / 128) % 32 -> pure shift/mask (shapes are fixed)
    const int   ch   = (row >> TV_R_SHIFT) & TV_C_MASK;
    const float lraw = lmbd_param[ch];
    const float lam  = fmaxf(lraw, 0.0f) + log1pf(expf(-fabsf(lraw)));

    const float* xrow = x + (size_t)row * TV_N;

    // ---- CDNA5 async global->LDS stage of this wave's row (1 KB) ----------
    // Each lane copies 32 contiguous bytes via two b128 async transfers.
    // LDS dst address = low 32 bits of the generic shared pointer (= LDS offset).
    unsigned lds0 = (unsigned)(uintptr_t)(&smem[0])
                  + (unsigned)(wv * TV_N * 4 + lane * 32);
    unsigned lds1 = lds0 + 16u;
    unsigned long long ga0 = (unsigned long long)(uintptr_t)xrow
                           + (unsigned long long)(lane * 32);
    unsigned long long ga1 = ga0 + 16ull;
    asm volatile(
        "global_load_async_to_lds_b128 %0, %1, off\n\t"
        "global_load_async_to_lds_b128 %2, %3, off\n\t"
        "s_wait_asynccnt 0"
        :
        : "v"(lds0), "v"(ga0), "v"(lds1), "v"(ga1)
        : "memory");

    // Pull the 8 per-lane elements into VGPRs (ds_load_b128 x2).
    const float4* sp = (const float4*)(&smem[wv * TV_N + lane * 8]);
    float4 a0 = sp[0];
    float4 a1 = sp[1];
    float xv[8] = {a0.x, a0.y, a0.z, a0.w, a1.x, a1.y, a1.z, a1.w};

    // Dual state: u (prev iterate) and y (momentum point), padded so that
    // logical index 255 (lane 31, k==7) is identically zero.
    float u[8], y[8];
#pragma unroll
    for (int k = 0; k < 8; ++k) { u[k] = 0.0f; y[k] = 0.0f; }
    float t = 1.0f;
    const bool lastLane = (lane == 31);

    for (int it = 0; it < TV_ITERS; ++it) {
        // w = x - D^T y ;  D^T y [i] = y[i-1] - y[i] with y[-1]=y[255]=0
        float yb = __shfl_up(y[7], 1);
        if (lane == 0) yb = 0.0f;
        float w[8];
        w[0] = xv[0] - (yb - y[0]);
#pragma unroll
        for (int k = 1; k < 8; ++k) w[k] = xv[k] - (y[k - 1] - y[k]);

        // grad = -D w ;  grad[i] = w[i] - w[i+1]
        float wn = __shfl_down(w[0], 1);
        float g[8];
#pragma unroll
        for (int k = 0; k < 7; ++k) g[k] = w[k] - w[k + 1];
        g[7] = w[7] - wn;               // lane31 slot is the padded index -> forced 0 below

        // momentum coefficient (data-independent, uniform across wave)
        float tn   = 0.5f * (1.0f + sqrtf(1.0f + 4.0f * t * t));
        float coef = (t - 1.0f) / tn;
        t = tn;

#pragma unroll
        for (int k = 0; k < 8; ++k) {
            float un = y[k] - 0.25f * g[k];
            un = fminf(lam, fmaxf(-lam, un));
            if (k == 7 && lastLane) un = 0.0f;   // pin u[255] == 0
            y[k] = un + coef * (un - u[k]);
            u[k] = un;
        }
    }

    // out = x - D^T u
    float ub = __shfl_up(u[7], 1);
    if (lane == 0) ub = 0.0f;
    float o[8];
    o[0] = xv[0] - (ub - u[0]);
#pragma unroll
    for (int k = 1; k < 8; ++k) o[k] = xv[k] - (u[k - 1] - u[k]);

    float4* op = (float4*)(out + (size_t)row * TV_N + lane * 8);
    op[0] = make_float4(o[0], o[1], o[2], o[3]);
    op[1] = make_float4(o[4], o[5], o[6], o[7]);
}

extern "C" void kernel_launch(void* const* d_in, const int* in_sizes, int n_in,
                              void* d_out, int out_size, void* d_ws, size_t ws_size,
                              hipStream_t stream)
{
    const float* x    = (const float*)d_in[0];
    const float* lmbd = (const float*)d_in[1];
    float*       out  = (float*)d_out;

    const int total = in_sizes[0];       // 8*32*128*256
    const int nrows = total / TV_N;      // 32768

    const int blocks = (nrows + TV_WAVES_PER_BLOCK - 1) / TV_WAVES_PER_BLOCK;
    tv1d_fista_kernel<<<blocks, TV_BLOCK, 0, stream>>>(x, lmbd, out, nrows);
}